// PointConv_9165460209727
// MI455X (gfx1250) — compile-verified
//
#include <hip/hip_runtime.h>

typedef __attribute__((ext_vector_type(2))) float v2f;
typedef __attribute__((ext_vector_type(8))) float v8f;

#define B_ 2
#define N_ 40000
#define K_ 16

static constexpr size_t R_ = (size_t)B_ * N_ * K_;   // 1,280,000 (b,n,k) rows
static constexpr size_t P_ = (size_t)B_ * N_;        // 80,000 points

// ---- workspace layout (in floats) ----
static constexpr size_t OFF_WNI = 0;                    // 12*R  weightNetInput
static constexpr size_t OFF_H1  = OFF_WNI + 12 * R_;    //  8*R  mlp1 pre-bn
static constexpr size_t OFF_H2  = OFF_H1  +  8 * R_;    //  8*R  mlp2 pre-bn
static constexpr size_t OFF_H3  = OFF_H2  +  8 * R_;    // 16*R  mlp3 pre-bn
static constexpr size_t OFF_Y1  = OFF_H3  + 16 * R_;    // 64*R  PE GEMM pre-bn
static constexpr size_t OFF_AM  = OFF_Y1  + 64 * R_;    // 1024*P einsum result
static constexpr size_t OFF_Y2  = OFF_AM  + 1024 * P_;  // 128*P final pre-bn
static constexpr size_t OFF_ST  = OFF_Y2  + 128 * P_;   // 1024 stats floats

// stats block per stage: [sum(nch) | sumsq(nch) | mean(nch) | istd(nch)]
#define ST_L1  0
#define ST_L2  32
#define ST_L3  64
#define ST_PE  128
#define ST_LIN 384

__device__ __forceinline__ v8f wmma4(v2f a, v2f b, v8f c) {
  // D(16x16 f32) = A(16x4 f32) * B(4x16 f32) + C
  return __builtin_amdgcn_wmma_f32_16x16x4_f32(false, a, false, b, (short)0, c,
                                               false, false);
}

// CDNA5 async global->LDS copy (16B per lane), tracked by ASYNCcnt.
__device__ __forceinline__ void async_copy_b128(void* lds_ptr, const void* gptr) {
  unsigned int  loff = (unsigned int)(size_t)lds_ptr;   // low 32 bits = LDS addr
  unsigned long long ga = (unsigned long long)(size_t)gptr;
  asm volatile("global_load_async_to_lds_b128 %0, %1, off"
               :: "v"(loff), "v"(ga)
               : "memory");
}
__device__ __forceinline__ void wait_async0() {
  asm volatile("s_wait_asynccnt 0x0" ::: "memory");
}

// ---------------- stats helpers ----------------
__global__ void k_zero(float* stats) {
  for (int i = threadIdx.x; i < 1024; i += 256) stats[i] = 0.f;
}

__global__ void k_fin(float* stats, int base, int nch, float cnt) {
  int c = threadIdx.x;
  if (c >= nch) return;
  float s = stats[base + c];
  float q = stats[base + nch + c];
  float m = s / cnt;
  float var = q / cnt - m * m;
  stats[base + 2 * nch + c] = m;
  stats[base + 3 * nch + c] = rsqrtf(var + 1e-5f);
}

// ---------------- geometry (weightNetInput) ----------------
__global__ void __launch_bounds__(256) k_geom(const float* __restrict__ xyz,
                                              const float* __restrict__ nrm,
                                              const int* __restrict__ nei,
                                              float* __restrict__ wni,
                                              float* __restrict__ out_tail) {
  size_t r = (size_t)blockIdx.x * 256 + threadIdx.x;   // grid covers R_ exactly
  size_t p = r / K_;                                   // point = b*N+n
  int b = (int)(p / N_);
  int idx = nei[r];
  size_t cb = p * 3;
  size_t gb = ((size_t)b * N_ + idx) * 3;

  float lx = xyz[gb + 0] - xyz[cb + 0];
  float ly = xyz[gb + 1] - xyz[cb + 1];
  float lz = xyz[gb + 2] - xyz[cb + 2];
  float gnx = nrm[gb + 0], gny = nrm[gb + 1], gnz = nrm[gb + 2];
  float nsx = nrm[cb + 0], nsy = nrm[cb + 1], nsz = nrm[cb + 2];

  float ln = sqrtf(lx * lx + ly * ly + lz * lz);
  float inv = 1.f / fmaxf(ln, 1e-12f);
  float rx = lx * inv, ry = ly * inv, rz = lz * inv;

  float proj = nsx * rx + nsy * ry + nsz * rz;
  float vx = nsx - proj * rx, vy = nsy - proj * ry, vz = nsz - proj * rz;
  float vn = sqrtf(vx * vx + vy * vy + vz * vz);
  float vinv = 1.f / fmaxf(vn, 1e-12f);
  vx *= vinv; vy *= vinv; vz *= vinv;

  float wx = ry * vz - rz * vy;
  float wy = rz * vx - rx * vz;
  float wz = rx * vy - ry * vx;
  float wn = sqrtf(wx * wx + wy * wy + wz * wz);
  float winv = 1.f / fmaxf(wn, 1e-12f);
  wx *= winv; wy *= winv; wz *= winv;

  float t1 = gnx * nsx + gny * nsy + gnz * nsz;
  float t2 = rx * nsx + ry * nsy + rz * nsz;
  float t3 = rx * gnx + ry * gny + rz * gnz;
  float t4 = lx * nsx + ly * nsy + lz * nsz;
  float t5 = t3;
  float t6 = gnx * vx + gny * vy + gnz * vz;
  float t7 = gnx * wx + gny * wy + gnz * wz;
  float cx = gny * nsz - gnz * nsy;
  float cy = gnz * nsx - gnx * nsz;
  float cz = gnx * nsy - gny * nsx;
  float t8 = lx * cx + ly * cy + lz * cz;
  float t9 = ln;

  float o[12] = {t1, t2, t3, t4, t5, t6, t7, t8, t9, lx, ly, lz};
  size_t base = r * 12;
#pragma unroll
  for (int j = 0; j < 12; ++j) {
    wni[base + j] = o[j];
    out_tail[base + j] = o[j];
  }
}

// ---------------- tiny weight-net MLP layers (scalar) ----------------
__global__ void __launch_bounds__(256) k_mlp1(const float* __restrict__ wni,
                                              const float* __restrict__ w1,
                                              const float* __restrict__ b1,
                                              float* __restrict__ h1,
                                              float* __restrict__ stats) {
  __shared__ float ssum[8], ssq[8];
  if (threadIdx.x < 8) { ssum[threadIdx.x] = 0.f; ssq[threadIdx.x] = 0.f; }
  __syncthreads();
  size_t r = (size_t)blockIdx.x * 256 + threadIdx.x;
  float a[12];
#pragma unroll
  for (int i = 0; i < 12; ++i) a[i] = wni[r * 12 + i];
#pragma unroll
  for (int j = 0; j < 8; ++j) {
    float v = b1[j];
#pragma unroll
    for (int i = 0; i < 12; ++i) v += a[i] * w1[i * 8 + j];
    h1[r * 8 + j] = v;
    atomicAdd(&ssum[j], v);
    atomicAdd(&ssq[j], v * v);
  }
  __syncthreads();
  if (threadIdx.x < 8) {
    atomicAdd(&stats[ST_L1 + threadIdx.x], ssum[threadIdx.x]);
    atomicAdd(&stats[ST_L1 + 8 + threadIdx.x], ssq[threadIdx.x]);
  }
}

__global__ void __launch_bounds__(256) k_mlp2(const float* __restrict__ h1,
                                              const float* __restrict__ w2,
                                              const float* __restrict__ b2,
                                              float* __restrict__ h2,
                                              float* __restrict__ stats) {
  __shared__ float ssum[8], ssq[8];
  if (threadIdx.x < 8) { ssum[threadIdx.x] = 0.f; ssq[threadIdx.x] = 0.f; }
  __syncthreads();
  size_t r = (size_t)blockIdx.x * 256 + threadIdx.x;
  float a[8];
#pragma unroll
  for (int i = 0; i < 8; ++i) {
    float m = stats[ST_L1 + 16 + i], s = stats[ST_L1 + 24 + i];
    a[i] = fmaxf((h1[r * 8 + i] - m) * s, 0.f);
  }
#pragma unroll
  for (int j = 0; j < 8; ++j) {
    float v = b2[j];
#pragma unroll
    for (int i = 0; i < 8; ++i) v += a[i] * w2[i * 8 + j];
    h2[r * 8 + j] = v;
    atomicAdd(&ssum[j], v);
    atomicAdd(&ssq[j], v * v);
  }
  __syncthreads();
  if (threadIdx.x < 8) {
    atomicAdd(&stats[ST_L2 + threadIdx.x], ssum[threadIdx.x]);
    atomicAdd(&stats[ST_L2 + 8 + threadIdx.x], ssq[threadIdx.x]);
  }
}

__global__ void __launch_bounds__(256) k_mlp3(const float* __restrict__ h2,
                                              const float* __restrict__ w3,
                                              const float* __restrict__ b3,
                                              float* __restrict__ h3,
                                              float* __restrict__ stats) {
  __shared__ float ssum[16], ssq[16];
  if (threadIdx.x < 16) { ssum[threadIdx.x] = 0.f; ssq[threadIdx.x] = 0.f; }
  __syncthreads();
  size_t r = (size_t)blockIdx.x * 256 + threadIdx.x;
  float a[8];
#pragma unroll
  for (int i = 0; i < 8; ++i) {
    float m = stats[ST_L2 + 16 + i], s = stats[ST_L2 + 24 + i];
    a[i] = fmaxf((h2[r * 8 + i] - m) * s, 0.f);
  }
#pragma unroll
  for (int j = 0; j < 16; ++j) {
    float v = b3[j];
#pragma unroll
    for (int i = 0; i < 8; ++i) v += a[i] * w3[i * 16 + j];
    h3[r * 16 + j] = v;
    atomicAdd(&ssum[j], v);
    atomicAdd(&ssq[j], v * v);
  }
  __syncthreads();
  if (threadIdx.x < 16) {
    atomicAdd(&stats[ST_L3 + threadIdx.x], ssum[threadIdx.x]);
    atomicAdd(&stats[ST_L3 + 16 + threadIdx.x], ssq[threadIdx.x]);
  }
}

// ---------------- PE branch: [feat(64)|wni(12)] @ w_pe(76x64), WMMA f32 ----
// w_pe staged k-pair-interleaved so each B fragment is one aligned ds_load_b64.
// A tile staged via CDNA5 async global->LDS B128 copies.
__global__ void __launch_bounds__(256) k_pe(const float* __restrict__ feats,
                                            const int* __restrict__ nei,
                                            const float* __restrict__ wni,
                                            const float* __restrict__ w_pe,
                                            const float* __restrict__ b_pe,
                                            float* __restrict__ y1,
                                            float* __restrict__ stats) {
  __shared__ __align__(16) float Wp[38 * 64 * 2];   // {W[2k][c], W[2k+1][c]}
  __shared__ float Bl[64];
  __shared__ __align__(16) float Al[8][16][76];
  __shared__ float ssum[64], ssq[64];

  int tid = threadIdx.x;
  int wave = tid >> 5, lane = tid & 31;
  for (int i = tid; i < 76 * 64; i += 256) {
    int k = i >> 6, c = i & 63;
    Wp[(((k >> 1) << 6) + c) * 2 + (k & 1)] = w_pe[i];
  }
  if (tid < 64) { Bl[tid] = b_pe[tid]; ssum[tid] = 0.f; ssq[tid] = 0.f; }

  size_t tile = (size_t)blockIdx.x * 8 + wave;   // == point index p
  size_t r0 = tile * 16;
  int b = (int)(tile / N_);

  // stage A tile asynchronously: 16 rows x [gathered feat(64) | wni(12)]
  for (int i = 0; i < 16; ++i) {
    size_t r = r0 + i;
    if (lane < 16) {
      int idx = nei[r];
      const float* fr = feats + ((size_t)b * N_ + idx) * 64;
      async_copy_b128(&Al[wave][i][lane * 4], fr + lane * 4);
    } else if (lane < 19) {
      int j = lane - 16;                         // 3 x 4 floats = 12
      async_copy_b128(&Al[wave][i][64 + j * 4], wni + r * 12 + j * 4);
    }
  }
  wait_async0();
  __syncthreads();

  int rowA = lane & 15, hi = lane >> 4, cb = lane & 15;
  v8f acc[4] = {};
  for (int kc = 0; kc < 19; ++kc) {           // 76 = 19 * 4
    int k0 = kc * 4 + 2 * hi;                 // even
    int kp = kc * 2 + hi;                     // pair index
    v2f a = *(const v2f*)&Al[wave][rowA][k0];
#pragma unroll
    for (int nt = 0; nt < 4; ++nt) {
      int col = nt * 16 + cb;
      v2f bf = *(const v2f*)&Wp[((kp << 6) + col) * 2];
      acc[nt] = wmma4(a, bf, acc[nt]);
    }
  }

  // epilogue: add bias, write pre-BN, accumulate channel stats
#pragma unroll
  for (int nt = 0; nt < 4; ++nt) {
    int col = nt * 16 + cb;
    float bs = Bl[col];
    float ls = 0.f, lq = 0.f;
#pragma unroll
    for (int i = 0; i < 8; ++i) {
      size_t row = r0 + i + 8 * hi;
      float v = acc[nt][i] + bs;
      y1[row * 64 + col] = v;
      ls += v; lq += v * v;
    }
    atomicAdd(&ssum[col], ls);
    atomicAdd(&ssq[col], lq);
  }
  __syncthreads();
  if (tid < 64) {
    atomicAdd(&stats[ST_PE + tid], ssum[tid]);
    atomicAdd(&stats[ST_PE + 64 + tid], ssq[tid]);
  }
}

// ---------------- einsum: per point A = pe^T(64x16) @ wt(16x16), WMMA ------
// Both operands staged k-pair-interleaved (fragments are ds_load_b64).
__global__ void __launch_bounds__(256) k_outer(const float* __restrict__ y1,
                                               const float* __restrict__ h3,
                                               const float* __restrict__ stats,
                                               float* __restrict__ am) {
  __shared__ __align__(16) float Pp[8][8 * 64 * 2];  // {pe[2k][c], pe[2k+1][c]}
  __shared__ __align__(16) float Tp[8][8 * 16 * 2];  // {wt[2k][w], wt[2k+1][w]}
  int tid = threadIdx.x, wave = tid >> 5, lane = tid & 31;
  size_t p = (size_t)blockIdx.x * 8 + wave;
  size_t rb = p * 16;

#pragma unroll
  for (int t = 0; t < 32; ++t) {
    int e = t * 32 + lane;           // 0..1023
    int k = e >> 6, c = e & 63;
    float m = stats[ST_PE + 128 + c], s = stats[ST_PE + 192 + c];
    float v = fmaxf((y1[(rb + k) * 64 + c] - m) * s, 0.f);
    Pp[wave][(((k >> 1) << 6) + c) * 2 + (k & 1)] = v;
  }
#pragma unroll
  for (int t = 0; t < 8; ++t) {
    int e = t * 32 + lane;           // 0..255
    int k = e >> 4, w = e & 15;
    float m = stats[ST_L3 + 32 + w], s = stats[ST_L3 + 48 + w];
    float v = fmaxf((h3[(rb + k) * 16 + w] - m) * s, 0.f);
    Tp[wave][(((k >> 1) << 4) + w) * 2 + (k & 1)] = v;
  }
  __syncthreads();

  int hi = lane >> 4, cb = lane & 15;
  v8f acc[4] = {};
#pragma unroll
  for (int kc = 0; kc < 4; ++kc) {   // K = 16 = 4 * 4
    int kp = kc * 2 + hi;
    v2f bf = *(const v2f*)&Tp[wave][((kp << 4) + cb) * 2];
#pragma unroll
    for (int ct = 0; ct < 4; ++ct) { // M = 64 = 4 tiles of 16 (c dimension)
      int c = ct * 16 + cb;
      v2f a = *(const v2f*)&Pp[wave][((kp << 6) + c) * 2];  // A'[c][k]=pe[k][c]
      acc[ct] = wmma4(a, bf, acc[ct]);
    }
  }
#pragma unroll
  for (int ct = 0; ct < 4; ++ct)
#pragma unroll
    for (int i = 0; i < 8; ++i) {
      int c = ct * 16 + i + 8 * hi;
      am[p * 1024 + (size_t)c * 16 + cb] = acc[ct][i];
    }
}

// ---------------- final linear: (P x 1024) @ (1024 x 128), WMMA ------------
// A and B tiles staged with async global->LDS B128 copies each K-chunk.
__global__ void __launch_bounds__(256) k_lin(const float* __restrict__ am,
                                             const float* __restrict__ w_lin,
                                             const float* __restrict__ b_lin,
                                             float* __restrict__ y2,
                                             float* __restrict__ stats) {
  __shared__ __align__(16) float Alds[16][68];  // padded (272B row, 16B mult)
  __shared__ __align__(16) float Blds[64][128];
  __shared__ float ssum[128], ssq[128];
  int tid = threadIdx.x, wave = tid >> 5, lane = tid & 31;
  if (tid < 128) { ssum[tid] = 0.f; ssq[tid] = 0.f; }
  size_t m0 = (size_t)blockIdx.x * 16;
  int hi = lane >> 4, cb = lane & 15, rowA = lane & 15;

  v8f acc = {};
  for (int kc = 0; kc < 16; ++kc) {  // K = 1024 = 16 chunks of 64
    size_t kb = (size_t)kc * 64;
    __syncthreads();                 // previous chunk fully consumed
    {                                // stage A: 16x64 floats, 1 B128/thread
      int e4 = tid * 4;
      int rr = e4 >> 6, cc = e4 & 63;
      async_copy_b128(&Alds[rr][cc], am + (m0 + rr) * 1024 + kb + cc);
    }
#pragma unroll
    for (int t = 0; t < 8; ++t) {    // stage B: 64x128 floats, 8 B128/thread
      int e4 = (t * 256 + tid) * 4;
      int rr = e4 >> 7, cc = e4 & 127;
      async_copy_b128(&Blds[rr][cc], w_lin + (kb + rr) * 128 + cc);
    }
    wait_async0();
    __syncthreads();
#pragma unroll
    for (int k4 = 0; k4 < 16; ++k4) {
      int k0 = k4 * 4 + 2 * hi;      // even
      v2f a = *(const v2f*)&Alds[rowA][k0];
      v2f bf;
      bf.x = Blds[k0][wave * 16 + cb];
      bf.y = Blds[k0 + 1][wave * 16 + cb];
      acc = wmma4(a, bf, acc);
    }
  }

  int col = wave * 16 + cb;
  float bs = b_lin[col];
  float ls = 0.f, lq = 0.f;
#pragma unroll
  for (int i = 0; i < 8; ++i) {
    size_t row = m0 + i + 8 * hi;
    float v = acc[i] + bs;
    y2[row * 128 + col] = v;
    ls += v; lq += v * v;
  }
  atomicAdd(&ssum[col], ls);
  atomicAdd(&ssq[col], lq);
  __syncthreads();
  if (tid < 128) {
    atomicAdd(&stats[ST_LIN + tid], ssum[tid]);
    atomicAdd(&stats[ST_LIN + 128 + tid], ssq[tid]);
  }
}

// ---------------- final BN + ReLU -> output head ----------------
__global__ void __launch_bounds__(256) k_out(const float* __restrict__ y2,
                                             const float* __restrict__ stats,
                                             float* __restrict__ out) {
  size_t i = (size_t)blockIdx.x * 256 + threadIdx.x;
  int c = (int)(i & 127);
  float m = stats[ST_LIN + 256 + c], s = stats[ST_LIN + 384 + c];
  out[i] = fmaxf((y2[i] - m) * s, 0.f);
}

extern "C" void kernel_launch(void* const* d_in, const int* in_sizes, int n_in,
                              void* d_out, int out_size, void* d_ws,
                              size_t ws_size, hipStream_t stream) {
  (void)in_sizes; (void)n_in; (void)out_size; (void)ws_size;
  const float* xyz   = (const float*)d_in[0];
  const float* feats = (const float*)d_in[1];
  const float* nrm   = (const float*)d_in[2];
  const int*   nei   = (const int*)d_in[3];
  const float* w_pe  = (const float*)d_in[4];
  const float* b_pe  = (const float*)d_in[5];
  const float* w1    = (const float*)d_in[6];
  const float* b1    = (const float*)d_in[7];
  const float* w2    = (const float*)d_in[8];
  const float* b2    = (const float*)d_in[9];
  const float* w3    = (const float*)d_in[10];
  const float* b3    = (const float*)d_in[11];
  const float* w_lin = (const float*)d_in[12];
  const float* b_lin = (const float*)d_in[13];

  float* out = (float*)d_out;
  float* ws  = (float*)d_ws;
  float* wni = ws + OFF_WNI;
  float* h1  = ws + OFF_H1;
  float* h2  = ws + OFF_H2;
  float* h3  = ws + OFF_H3;
  float* y1  = ws + OFF_Y1;
  float* am  = ws + OFF_AM;
  float* y2  = ws + OFF_Y2;
  float* st  = ws + OFF_ST;

  float* out_head = out;                      // new_feat: 128 * P floats
  float* out_tail = out + 128 * P_;           // weightNetInput: 12 * R floats

  int gR = (int)(R_ / 256);                   // 5000
  k_zero<<<1, 256, 0, stream>>>(st);
  k_geom<<<gR, 256, 0, stream>>>(xyz, nrm, nei, wni, out_tail);

  k_mlp1<<<gR, 256, 0, stream>>>(wni, w1, b1, h1, st);
  k_fin<<<1, 64, 0, stream>>>(st, ST_L1, 8, (float)R_);
  k_mlp2<<<gR, 256, 0, stream>>>(h1, w2, b2, h2, st);
  k_fin<<<1, 64, 0, stream>>>(st, ST_L2, 8, (float)R_);
  k_mlp3<<<gR, 256, 0, stream>>>(h2, w3, b3, h3, st);
  k_fin<<<1, 64, 0, stream>>>(st, ST_L3, 16, (float)R_);

  k_pe<<<(int)(R_ / 128), 256, 0, stream>>>(feats, nei, wni, w_pe, b_pe, y1, st);
  k_fin<<<1, 64, 0, stream>>>(st, ST_PE, 64, (float)R_);

  k_outer<<<(int)(P_ / 8), 256, 0, stream>>>(y1, h3, st, am);

  k_lin<<<(int)(P_ / 16), 256, 0, stream>>>(am, w_lin, b_lin, y2, st);
  k_fin<<<1, 128, 0, stream>>>(st, ST_LIN, 128, (float)P_);

  k_out<<<(int)(128 * P_ / 256), 256, 0, stream>>>(y2, st, out_head);
}